// TrainableFrangiFeatureExtractor_79654463471931
// MI455X (gfx1250) — compile-verified
//
#include <hip/hip_runtime.h>
#include <hip/hip_bf16.h>

typedef __attribute__((ext_vector_type(2))) float v2f;
typedef __attribute__((ext_vector_type(8))) float v8f;

#if __has_builtin(__builtin_amdgcn_wmma_f32_16x16x4_f32)
#define WMMA_F32_K4(acc, a, b) \
  (acc) = __builtin_amdgcn_wmma_f32_16x16x4_f32(false, (a), false, (b), (short)0, (acc), false, false)
#else
#define WMMA_F32_K4(acc, a, b) \
  asm volatile("v_wmma_f32_16x16x4_f32 %0, %1, %2, %0" : "+v"(acc) : "v"(a), "v"(b))
#endif

#define VOL  4194304   // 2*1*128*128*128
#define NX   128
#define DSTR 16384     // d stride
#define HSTR 128       // h stride

// ---------------------------------------------------------------------------
// Prep: normalized 1D Gaussian taps from the 3D kernel's center line.
// kern3d[3][3][t] = e_t / s1^3, so w[t] = kern[168+t] / sum_t(kern[168+t]).
// ---------------------------------------------------------------------------
__global__ void frangi_prep_weights(const float* __restrict__ kern, float* __restrict__ w7) {
    if (threadIdx.x == 0) {
        float v[7], s = 0.0f;
        #pragma unroll
        for (int t = 0; t < 7; ++t) { v[t] = kern[168 + t]; s += v[t]; }
        float inv = 1.0f / s;
        #pragma unroll
        for (int t = 0; t < 7; ++t) w7[t] = v[t] * inv;
    }
}

// Build 24x16 Toeplitz table T[k][c] = w[k-c] (0<=k-c<7 else 0) in LDS.
__device__ __forceinline__ void build_toeplitz(float* tp, const float* __restrict__ w7, int tid) {
    for (int i = tid; i < 24 * 16; i += 128) {
        int k = i >> 4, c = i & 15;
        int t = k - c;
        tp[i] = (t >= 0 && t < 7) ? w7[t] : 0.0f;
    }
}

// ---------------------------------------------------------------------------
// Pass 1: blur along W. Block (128 thr = 4 waves) owns a 16-row x 128-col slab.
// Slab staged in LDS with zero-padded halo; each wave does 2 WMMA tiles:
//   O(16x16) = A(image 16x24) * T(Toeplitz 24x16), 6 K=4 chunks.
// ---------------------------------------------------------------------------
#define WSTRIDE 137   // 128 + 8 halo + 1 bank-skew
__global__ __launch_bounds__(128) void frangi_blur_w(const float* __restrict__ in,
                                                     float* __restrict__ out,
                                                     const float* __restrict__ w7) {
    __shared__ float tile[16 * WSTRIDE];
    __shared__ float tp[24 * 16];

    const int tid = threadIdx.x;
    const int rb  = blockIdx.x * 16;        // 2048 row-slabs over B*D*H

    build_toeplitz(tp, w7, tid);
    // zero halo: lds cols 0..2 (w=-3..-1) and 131..135 (w=128..132)
    for (int i = tid; i < 16 * 8; i += 128) {
        int r = i >> 3, p = i & 7;
        tile[r * WSTRIDE + ((p < 3) ? p : (128 + p))] = 0.0f;
    }
    // coalesced slab load: lds col (w+3)
    for (int i = tid; i < 16 * 128; i += 128) {
        int r = i >> 7, cc = i & 127;
        tile[r * WSTRIDE + 3 + cc] = in[(size_t)(rb + r) * NX + cc];
    }
    __syncthreads();

    const int L  = tid & 31;
    const int wv = tid >> 5;
    const int c  = L & 15;            // N col (B/D) and A row M
    const int kb = 2 * (L >> 4);      // K sub-offset per lane half

    for (int tx = wv; tx < 8; tx += 4) {
        const int w0 = tx * 16;
        v8f acc = {};
        #pragma unroll
        for (int j = 0; j < 6; ++j) {
            const int k0 = 4 * j + kb;
            v2f a, b;
            a.x = tile[c * WSTRIDE + w0 + k0];       // input col w0-3+k0 -> lds idx +3
            a.y = tile[c * WSTRIDE + w0 + k0 + 1];
            b.x = tp[k0 * 16 + c];
            b.y = tp[(k0 + 1) * 16 + c];
            WMMA_F32_K4(acc, a, b);
        }
        float* ob = out + (size_t)rb * NX + w0 + c;
        const int mb = 8 * (L >> 4);
        #pragma unroll
        for (int v = 0; v < 8; ++v)
            ob[(size_t)(mb + v) * NX] = acc[v];
    }
}

// ---------------------------------------------------------------------------
// Passes 2/3: blur along strided axis (H or D); blur axis is M:
//   O(16 axis x 16 w) = T(16x24) * I(24x16). 24 axis-rows staged in LDS.
//   H pass: innerStride=16384 (fixed d), axisStride=128
//   D pass: innerStride=128   (fixed h), axisStride=16384
// ---------------------------------------------------------------------------
#define ASTRIDE 136   // 128 + 8 skew => lane halves hit disjoint bank windows
__global__ __launch_bounds__(128) void frangi_blur_axis(const float* __restrict__ in,
                                                        float* __restrict__ out,
                                                        const float* __restrict__ w7,
                                                        int innerStride, int axisStride) {
    __shared__ float tile[24 * ASTRIDE];
    __shared__ float tp[24 * 16];

    const int tid = threadIdx.x;
    const int a0  = blockIdx.x * 16;        // axis tile
    const int z   = blockIdx.y;             // 0..255 plane
    const size_t base = (size_t)(z >> 7) * 2097152u + (size_t)(z & 127) * (size_t)innerStride;

    build_toeplitz(tp, w7, tid);
    // stage 24 axis-rows x 128 cols (zero rows outside [0,128))
    for (int i = tid; i < 24 * 128; i += 128) {
        int k = i >> 7, cc = i & 127;
        int ai = a0 - 3 + k;
        tile[k * ASTRIDE + cc] =
            (ai >= 0 && ai < NX) ? in[base + (size_t)ai * axisStride + cc] : 0.0f;
    }
    __syncthreads();

    const int L  = tid & 31;
    const int wv = tid >> 5;
    const int c  = L & 15;            // N (w col) and A row M
    const int kb = 2 * (L >> 4);

    for (int tx = wv; tx < 8; tx += 4) {
        const int w0 = tx * 16;
        v8f acc = {};
        #pragma unroll
        for (int j = 0; j < 6; ++j) {
            const int k0 = 4 * j + kb;
            v2f a, b;
            a.x = tp[k0 * 16 + c];                    // Toeplitz in A: T[m][k]=w[k-m]
            a.y = tp[(k0 + 1) * 16 + c];
            b.x = tile[k0 * ASTRIDE + w0 + c];
            b.y = tile[(k0 + 1) * ASTRIDE + w0 + c];
            WMMA_F32_K4(acc, a, b);
        }
        const int mb = 8 * (L >> 4);
        #pragma unroll
        for (int v = 0; v < 8; ++v)
            out[base + (size_t)(a0 + mb + v) * axisStride + w0 + c] = acc[v];
    }
}

// ---------------------------------------------------------------------------
// Frangi stage: Hessian (torch.gradient semantics) + trig 3x3 eigensolve.
// Interior voxels (91%) use closed-form branch-free stencils.
// ---------------------------------------------------------------------------
__device__ __forceinline__ float smp(const float* s, int b, int d, int h, int w) {
    return s[((((size_t)b * NX + d) * NX + h) << 7) + w];
}

template <int AX>
__device__ __forceinline__ float sampAt(const float* s, int b, int d, int h, int w, int i) {
    if (AX == 0) d = i; else if (AX == 1) h = i; else w = i;
    return smp(s, b, d, h, w);
}

template <int AX>
__device__ __forceinline__ float grad1(const float* s, int b, int d, int h, int w) {
    const int i = (AX == 0) ? d : (AX == 1) ? h : w;
    if (i == 0)      return sampAt<AX>(s, b, d, h, w, 1)   - sampAt<AX>(s, b, d, h, w, 0);
    if (i == NX - 1) return sampAt<AX>(s, b, d, h, w, 127) - sampAt<AX>(s, b, d, h, w, 126);
    return 0.5f * (sampAt<AX>(s, b, d, h, w, i + 1) - sampAt<AX>(s, b, d, h, w, i - 1));
}

template <int AXA, int AXB>
__device__ __forceinline__ float grad2(const float* s, int b, int d, int h, int w) {
    const int j = (AXB == 0) ? d : (AXB == 1) ? h : w;
    auto g = [&](int jj) {
        int dd = d, hh = h, ww = w;
        if (AXB == 0) dd = jj; else if (AXB == 1) hh = jj; else ww = jj;
        return grad1<AXA>(s, b, dd, hh, ww);
    };
    if (j == 0)      return g(1)   - g(0);
    if (j == NX - 1) return g(127) - g(126);
    return 0.5f * (g(j + 1) - g(j - 1));
}

__global__ __launch_bounds__(256) void frangi_vesselness(const float* __restrict__ sm,
                                                         float* __restrict__ out,
                                                         const float* __restrict__ pAlpha,
                                                         const float* __restrict__ pBeta,
                                                         const float* __restrict__ pC) {
    const int idx = blockIdx.x * 256 + threadIdx.x;
    const int w = idx & 127, h = (idx >> 7) & 127, d = (idx >> 14) & 127, b = idx >> 21;

    float Hxx, Hyy, Hzz, Hxy, Hxz, Hyz;
    const bool interior = (d >= 2) & (d <= 125) & (h >= 2) & (h <= 125) & (w >= 2) & (w <= 125);
    if (interior) {
        const float* p = sm + idx;
        const float f0 = p[0];
        Hxx = 0.25f * (p[ 2 * DSTR] - 2.0f * f0 + p[-2 * DSTR]);
        Hyy = 0.25f * (p[ 2 * HSTR] - 2.0f * f0 + p[-2 * HSTR]);
        Hzz = 0.25f * (p[ 2       ] - 2.0f * f0 + p[-2       ]);
        Hxy = 0.25f * (p[ DSTR + HSTR] - p[ DSTR - HSTR] - p[-DSTR + HSTR] + p[-DSTR - HSTR]);
        Hxz = 0.25f * (p[ DSTR + 1   ] - p[ DSTR - 1   ] - p[-DSTR + 1   ] + p[-DSTR - 1   ]);
        Hyz = 0.25f * (p[ HSTR + 1   ] - p[ HSTR - 1   ] - p[-HSTR + 1   ] + p[-HSTR - 1   ]);
    } else {
        Hxx = grad2<0, 0>(sm, b, d, h, w);
        Hyy = grad2<1, 1>(sm, b, d, h, w);
        Hzz = grad2<2, 2>(sm, b, d, h, w);
        Hxy = grad2<0, 1>(sm, b, d, h, w);
        Hxz = grad2<0, 2>(sm, b, d, h, w);
        Hyz = grad2<1, 2>(sm, b, d, h, w);
    }

    // Trig eigensolve of symmetric 3x3 (matches reference arithmetic).
    const float q   = (Hxx + Hyy + Hzz) * (1.0f / 3.0f);
    const float cxx = Hxx - q, cyy = Hyy - q, czz = Hzz - q;
    const float pn  = sqrtf((cxx * cxx + cyy * cyy + czz * czz +
                             2.0f * (Hxy * Hxy + Hxz * Hxz + Hyz * Hyz)) * (1.0f / 6.0f));
    const float inv = 1.0f / (pn + 1e-10f);
    const float bxx = cxx * inv, byy = cyy * inv, bzz = czz * inv;
    const float bxy = Hxy * inv, bxz = Hxz * inv, byz = Hyz * inv;
    const float detB = bxx * (byy * bzz - byz * byz)
                     - bxy * (bxy * bzz - byz * bxz)
                     + bxz * (bxy * byz - byy * bxz);
    const float th = acosf(fminf(fmaxf(detB * 0.5f, -1.0f), 1.0f)) * (1.0f / 3.0f);
    const float e0 = 2.0f * cosf(th) * pn + q;
    const float e1 = 2.0f * cosf(th + 2.0943951023931953f) * pn + q;
    const float e2 = 2.0f * cosf(th + 4.1887902047863905f) * pn + q;

    const float lo = fminf(e0, e1), hi = fmaxf(e0, e1);
    const float l1 = fminf(lo, e2);
    const float l3 = fmaxf(hi, e2);
    const float l2 = fminf(fmaxf(lo, e2), hi);

    const float al = *pAlpha, be = *pBeta, cc = *pC;
    const float RA = fabsf(l2 / (l3 + 1e-10f));
    const float RB = fabsf(l1 / sqrtf(fabsf(l2 * l3) + 1e-10f));
    const float S2 = l1 * l1 + l2 * l2 + l3 * l3;

    const float v = (1.0f - __expf(-(RA * RA) / (2.0f * al * al)))
                  * __expf(-(RB * RB) / (2.0f * be * be))
                  * (1.0f - __expf(-S2 / (2.0f * cc * cc)));

    out[idx] = (l2 < 0.0f && l3 < 0.0f) ? v : 0.0f;
}

// ---------------------------------------------------------------------------
extern "C" void kernel_launch(void* const* d_in, const int* in_sizes, int n_in,
                              void* d_out, int out_size, void* d_ws, size_t ws_size,
                              hipStream_t stream) {
    const float* x     = (const float*)d_in[0];
    const float* kern  = (const float*)d_in[1];
    const float* alpha = (const float*)d_in[2];
    const float* beta  = (const float*)d_in[3];
    const float* cpar  = (const float*)d_in[4];
    float* outp = (float*)d_out;

    float* w7   = (float*)d_ws;
    float* bufA = (float*)((char*)d_ws + 256);
    float* bufB = bufA + VOL;

    frangi_prep_weights<<<1, 32, 0, stream>>>(kern, w7);

    // Pass 1: blur along W (contiguous). B*D*H = 32768 rows -> 2048 slabs.
    frangi_blur_w<<<2048, 128, 0, stream>>>(x, bufA, w7);
    // Pass 2: blur along H (fixed b,d plane).
    frangi_blur_axis<<<dim3(8, 256), 128, 0, stream>>>(bufA, bufB, w7, DSTR, HSTR);
    // Pass 3: blur along D (fixed b,h plane).
    frangi_blur_axis<<<dim3(8, 256), 128, 0, stream>>>(bufB, bufA, w7, HSTR, DSTR);
    // Pass 4: Hessian + eigen + vesselness.
    frangi_vesselness<<<VOL / 256, 256, 0, stream>>>(bufA, outp, alpha, beta, cpar);
}